// Model_12283606468269
// MI455X (gfx1250) — compile-verified
//
#include <hip/hip_runtime.h>

#define N_NODES 50000
#define N_EDGES 800000
#define DIN     96
#define D       128
#define B_GR    512
#define MPNN_STEPS 4
#define S2S_STEPS  3

typedef __attribute__((ext_vector_type(2))) float v2f;
typedef __attribute__((ext_vector_type(8))) float v8f;

__device__ __forceinline__ float sigmoidf(float x) { return 1.0f / (1.0f + expf(-x)); }

__device__ __forceinline__ void atomicMaxFloat(float* addr, float val) {
    unsigned int* ua = (unsigned int*)addr;
    unsigned int old = *ua;
    while (__uint_as_float(old) < val) {
        unsigned int assumed = old;
        old = atomicCAS(ua, assumed, __float_as_uint(val));
        if (old == assumed) break;
    }
}

// ---------------------------------------------------------------------------
// Input layer: h = relu(x @ W_in^T + b_in).  One block = 16 nodes x 128 cols.
// 8 waves/block, each wave computes a 16x16 tile via 24x V_WMMA_F32_16X16X4_F32.
// ---------------------------------------------------------------------------
__global__ void __launch_bounds__(256)
gemm_in_kernel(const float* __restrict__ x, const float* __restrict__ Win,
               const float* __restrict__ bin, float* __restrict__ h)
{
    __shared__ float xs[16 * 100];     // 16 node rows, padded stride 100
    __shared__ float wl[128 * 100];    // all of W_in, padded stride 100

    const int tid     = threadIdx.x;
    const int rowbase = blockIdx.x * 16;

    for (int i = tid; i < 16 * DIN; i += 256)
        xs[(i / DIN) * 100 + (i % DIN)] = x[(size_t)rowbase * DIN + i];
    for (int i = tid; i < D * DIN; i += 256)
        wl[(i / DIN) * 100 + (i % DIN)] = Win[i];
    __syncthreads();

    const int lane = tid & 31;
    const int wave = tid >> 5;
    const int mr   = lane & 15;        // row (A) / col (B,C) index within tile
    const int half = lane >> 4;        // K-half selector per ISA layout

    const float* xrow = &xs[mr * 100 + 2 * half];
    const float* wrow = &wl[(wave * 16 + mr) * 100 + 2 * half];

    v8f acc = {0.f, 0.f, 0.f, 0.f, 0.f, 0.f, 0.f, 0.f};
#pragma unroll
    for (int k0 = 0; k0 < DIN; k0 += 4) {
        v2f a_; a_.x = xrow[k0]; a_.y = xrow[k0 + 1];
        v2f b_; b_.x = wrow[k0]; b_.y = wrow[k0 + 1];
        acc = __builtin_amdgcn_wmma_f32_16x16x4_f32(false, a_, false, b_,
                                                    (short)0, acc, false, false);
    }

    const int   col  = wave * 16 + mr;
    const float bias = bin[col];
#pragma unroll
    for (int j = 0; j < 8; ++j) {
        float v = acc[j] + bias;
        h[(size_t)(rowbase + j + 8 * half) * D + col] = v > 0.f ? v : 0.f;
    }
}

// ---------------------------------------------------------------------------
__global__ void zerof_kernel(float* __restrict__ p, int n)
{
    int t = blockIdx.x * blockDim.x + threadIdx.x;
    if (t < n) p[t] = 0.f;
}

__global__ void deg_kernel(const int* __restrict__ ei, float* __restrict__ deg)
{
    int e = blockIdx.x * blockDim.x + threadIdx.x;
    if (e >= N_EDGES) return;
    atomicAdd(&deg[ei[N_EDGES + e]], 1.0f);
}

// wave per edge: 32 lanes each move one float4 of the 128-float row
__global__ void scatter_kernel(const float* __restrict__ h,
                               const int* __restrict__ ei,
                               float* __restrict__ msg)
{
    int t = blockIdx.x * blockDim.x + threadIdx.x;
    if (t >= N_EDGES * 32) return;
    int e = t >> 5, c = t & 31;
    int src = ei[e];
    int dst = ei[N_EDGES + e];
    float4 v = ((const float4*)(h + (size_t)src * D))[c];
    float* m = msg + (size_t)dst * D + c * 4;
    atomicAdd(m + 0, v.x); atomicAdd(m + 1, v.y);
    atomicAdd(m + 2, v.z); atomicAdd(m + 3, v.w);
}

__global__ void update_kernel(float* __restrict__ h, const float* __restrict__ msg,
                              const float* __restrict__ deg)
{
    int i = blockIdx.x * blockDim.x + threadIdx.x;   // float4 index
    if (i >= N_NODES * (D / 4)) return;
    int   node = i / (D / 4);
    float dinv = 1.0f / fmaxf(deg[node], 1.0f);
    float4 hv = ((const float4*)h)[i];
    float4 mv = ((const float4*)msg)[i];
    hv.x = (hv.x + mv.x * dinv) * 0.5f;
    hv.y = (hv.y + mv.y * dinv) * 0.5f;
    hv.z = (hv.z + mv.z * dinv) * 0.5f;
    hv.w = (hv.w + mv.w * dinv) * 0.5f;
    ((float4*)h)[i] = hv;
}

// ---------------------------------------------------------------------------
// Set2Set LSTM gates via WMMA:  gates = [q_star | hh] @ [W_ih | W_hh]^T + biases
// Z = [B x 384] staged tile-wise in LDS; K = 384 -> 96 WMMA steps per tile.
// Grid: (32 row tiles, 4 col groups); 8 waves/block, 16x16 tile per wave.
// ---------------------------------------------------------------------------
__global__ void __launch_bounds__(256)
gates_wmma_kernel(const float* __restrict__ qs, const float* __restrict__ hh,
                  const float* __restrict__ Wih, const float* __restrict__ Whh,
                  const float* __restrict__ bih, const float* __restrict__ bhh,
                  float* __restrict__ gates)
{
    __shared__ float zl[16 * 388];     // 16 graph rows x 384 (padded stride 388)

    const int tid     = threadIdx.x;
    const int rowbase = blockIdx.x * 16;

    for (int i = tid; i < 16 * 384; i += 256) {
        int m = i / 384, c = i % 384;
        float v = (c < 2 * D) ? qs[(size_t)(rowbase + m) * 2 * D + c]
                              : hh[(size_t)(rowbase + m) * D + (c - 2 * D)];
        zl[m * 388 + c] = v;
    }
    __syncthreads();

    const int lane = tid & 31;
    const int wave = tid >> 5;
    const int mr   = lane & 15;
    const int half = lane >> 4;
    const int col  = (blockIdx.y * 8 + wave) * 16 + mr;   // gate index j

    const float* zrow = &zl[mr * 388 + 2 * half];
    const float* wi   = Wih + (size_t)col * (2 * D) + 2 * half;
    const float* wh   = Whh + (size_t)col * D + 2 * half;

    v8f acc = {0.f, 0.f, 0.f, 0.f, 0.f, 0.f, 0.f, 0.f};
#pragma unroll
    for (int k0 = 0; k0 < 2 * D; k0 += 4) {               // q_star part (K=256)
        v2f a_; a_.x = zrow[k0]; a_.y = zrow[k0 + 1];
        v2f b_; b_.x = wi[k0];   b_.y = wi[k0 + 1];
        acc = __builtin_amdgcn_wmma_f32_16x16x4_f32(false, a_, false, b_,
                                                    (short)0, acc, false, false);
    }
#pragma unroll
    for (int k0 = 0; k0 < D; k0 += 4) {                   // hh part (K=128)
        v2f a_; a_.x = zrow[2 * D + k0]; a_.y = zrow[2 * D + k0 + 1];
        v2f b_; b_.x = wh[k0];           b_.y = wh[k0 + 1];
        acc = __builtin_amdgcn_wmma_f32_16x16x4_f32(false, a_, false, b_,
                                                    (short)0, acc, false, false);
    }

    const float bias = bih[col] + bhh[col];
#pragma unroll
    for (int j = 0; j < 8; ++j)
        gates[(size_t)(rowbase + j + 8 * half) * (4 * D) + col] = acc[j] + bias;
}

__global__ void lstm_kernel(const float* __restrict__ gates, float* __restrict__ cc,
                            float* __restrict__ hh, float* __restrict__ qs)
{
    int t = blockIdx.x * blockDim.x + threadIdx.x;
    if (t >= B_GR * D) return;
    int b = t >> 7, j = t & 127;
    const float* g = gates + (size_t)b * 4 * D;
    float ig = sigmoidf(g[j]);
    float fg = sigmoidf(g[D + j]);
    float gg = tanhf(g[2 * D + j]);
    float og = sigmoidf(g[3 * D + j]);
    float c  = fg * cc[t] + ig * gg;
    cc[t] = c;
    float hn = og * tanhf(c);
    hh[t] = hn;
    qs[(size_t)b * 2 * D + j] = hn;            // q part of q_star
}

__global__ void attn_init_kernel(float* __restrict__ qs, float* __restrict__ emax,
                                 float* __restrict__ asum)
{
    int t = blockIdx.x * blockDim.x + threadIdx.x;
    if (t >= B_GR * D) return;
    int b = t >> 7, j = t & 127;
    qs[(size_t)b * 2 * D + D + j] = 0.f;       // zero r accumulator
    if (j == 0) { emax[b] = -3.402823466e38f; asum[b] = 0.f; }
}

// wave per node: e[i] = dot(h[i], q[batch[i]]); atomicMax into per-graph max
__global__ void e_kernel(const float* __restrict__ h, const float* __restrict__ hh,
                         const int* __restrict__ batch, float* __restrict__ ev,
                         float* __restrict__ emax)
{
    int t = blockIdx.x * blockDim.x + threadIdx.x;
    if (t >= N_NODES * 32) return;
    int node = t >> 5, lane = t & 31;
    int b = batch[node];
    float4 hv = ((const float4*)(h  + (size_t)node * D))[lane];
    float4 qv = ((const float4*)(hh + (size_t)b    * D))[lane];
    float s = hv.x * qv.x + hv.y * qv.y + hv.z * qv.z + hv.w * qv.w;
    for (int off = 16; off > 0; off >>= 1) s += __shfl_xor(s, off, 32);
    if (lane == 0) { ev[node] = s; atomicMaxFloat(&emax[b], s); }
}

__global__ void a_kernel(const float* __restrict__ ev, const float* __restrict__ emax,
                         const int* __restrict__ batch, float* __restrict__ av,
                         float* __restrict__ asum)
{
    int t = blockIdx.x * blockDim.x + threadIdx.x;
    if (t >= N_NODES) return;
    int b = batch[t];
    float va = expf(ev[t] - emax[b]);
    av[t] = va;
    atomicAdd(&asum[b], va);
}

// wave per node: r[b] += (a_i / a_sum[b]) * h[i]  (into second half of q_star)
__global__ void r_kernel(const float* __restrict__ h, const float* __restrict__ av,
                         const float* __restrict__ asum, const int* __restrict__ batch,
                         float* __restrict__ qs)
{
    int t = blockIdx.x * blockDim.x + threadIdx.x;
    if (t >= N_NODES * 32) return;
    int node = t >> 5, lane = t & 31;
    int b = batch[node];
    float w = av[node] / (asum[b] + 1e-16f);
    float4 hv = ((const float4*)(h + (size_t)node * D))[lane];
    float* r = qs + (size_t)b * 2 * D + D + lane * 4;
    atomicAdd(r + 0, w * hv.x); atomicAdd(r + 1, w * hv.y);
    atomicAdd(r + 2, w * hv.z); atomicAdd(r + 3, w * hv.w);
}

__global__ void pred_kernel(const float* __restrict__ qs, const float* __restrict__ Wp,
                            const float* __restrict__ bp, float* __restrict__ out)
{
    int t = blockIdx.x * blockDim.x + threadIdx.x;
    if (t >= B_GR * 32) return;
    int b = t >> 5, lane = t & 31;
    const float* q = qs + (size_t)b * 2 * D;
    float s = 0.f;
#pragma unroll
    for (int k = 0; k < 8; ++k) s += q[lane * 8 + k] * Wp[lane * 8 + k];
    for (int off = 16; off > 0; off >>= 1) s += __shfl_xor(s, off, 32);
    if (lane == 0) out[b] = s + bp[0];
}

// ---------------------------------------------------------------------------
extern "C" void kernel_launch(void* const* d_in, const int* in_sizes, int n_in,
                              void* d_out, int out_size, void* d_ws, size_t ws_size,
                              hipStream_t stream)
{
    const float* x     = (const float*)d_in[0];
    const int*   ei    = (const int*)  d_in[1];   // [2,E] int32
    const int*   batch = (const int*)  d_in[2];   // [N]   int32
    const float* Win   = (const float*)d_in[3];
    const float* bin   = (const float*)d_in[4];
    const float* Wih   = (const float*)d_in[5];
    const float* Whh   = (const float*)d_in[6];
    const float* bih   = (const float*)d_in[7];
    const float* bhh   = (const float*)d_in[8];
    const float* Wp    = (const float*)d_in[9];
    const float* bp    = (const float*)d_in[10];
    float* out = (float*)d_out;

    float* ws    = (float*)d_ws;
    float* h     = ws;                               // N*D
    float* msg   = h    + (size_t)N_NODES * D;       // N*D
    float* deg   = msg  + (size_t)N_NODES * D;       // N
    float* ev    = deg  + N_NODES;                   // N
    float* av    = ev   + N_NODES;                   // N
    float* emax  = av   + N_NODES;                   // B
    float* asum  = emax + B_GR;                      // B
    float* hh    = asum + B_GR;                      // B*D
    float* cc    = hh   + (size_t)B_GR * D;          // B*D
    float* qs    = cc   + (size_t)B_GR * D;          // B*2D
    float* gates = qs   + (size_t)B_GR * 2 * D;      // B*4D

    const dim3 blk(256);

    // 1) input layer (WMMA f32)
    gemm_in_kernel<<<N_NODES / 16, blk, 0, stream>>>(x, Win, bin, h);

    // 2) in-degree
    zerof_kernel<<<(N_NODES + 255) / 256, blk, 0, stream>>>(deg, N_NODES);
    deg_kernel<<<(N_EDGES + 255) / 256, blk, 0, stream>>>(ei, deg);

    // 3) MPNN steps
    for (int s = 0; s < MPNN_STEPS; ++s) {
        zerof_kernel<<<(N_NODES * D + 255) / 256, blk, 0, stream>>>(msg, N_NODES * D);
        scatter_kernel<<<(N_EDGES * 32 + 255) / 256, blk, 0, stream>>>(h, ei, msg);
        update_kernel<<<(N_NODES * (D / 4) + 255) / 256, blk, 0, stream>>>(h, msg, deg);
    }

    // 4) Set2Set (hh, cc, qs are contiguous: B*4D floats starting at hh)
    zerof_kernel<<<(B_GR * 4 * D + 255) / 256, blk, 0, stream>>>(hh, B_GR * 4 * D);
    for (int s = 0; s < S2S_STEPS; ++s) {
        gates_wmma_kernel<<<dim3(B_GR / 16, 4), blk, 0, stream>>>(qs, hh, Wih, Whh,
                                                                  bih, bhh, gates);
        lstm_kernel<<<(B_GR * D + 255) / 256, blk, 0, stream>>>(gates, cc, hh, qs);
        attn_init_kernel<<<(B_GR * D + 255) / 256, blk, 0, stream>>>(qs, emax, asum);
        e_kernel<<<(N_NODES * 32 + 255) / 256, blk, 0, stream>>>(h, hh, batch, ev, emax);
        a_kernel<<<(N_NODES + 255) / 256, blk, 0, stream>>>(ev, emax, batch, av, asum);
        r_kernel<<<(N_NODES * 32 + 255) / 256, blk, 0, stream>>>(h, av, asum, batch, qs);
    }

    // 5) prediction head
    pred_kernel<<<(B_GR * 32 + 255) / 256, blk, 0, stream>>>(qs, Wp, bp, out);
}